// SourceModuleHnNSF_65936337928763
// MI455X (gfx1250) — compile-verified
//
#include <hip/hip_runtime.h>

// ---------------------------------------------------------------------------
// Problem constants (from setup_inputs): B=16, F=1000 frames, upp=300.
// ---------------------------------------------------------------------------
#define BATCH      16
#define FRAMES     1000
#define UPP        300
#define SAMPLES    (FRAMES * UPP)          // 300000 per batch
#define TOTAL      (BATCH * SAMPLES)       // 4,800,000
#define FSTRIDE    1024                    // padded frame stride in workspace
#define NHARM      9                       // HARMONIC_NUM + 1
#define SINE_AMP   0.1f

typedef float v2f __attribute__((ext_vector_type(2)));
typedef float v4f __attribute__((ext_vector_type(4)));
typedef float v8f __attribute__((ext_vector_type(8)));

// mod-1 (hardware v_fract_f32: x - floor(x), valid for negatives too)
__device__ __forceinline__ float frac1(float x) { return __builtin_amdgcn_fractf(x); }

// hardware sin: v_sin_f32 computes sin(2*pi*x) -> feed phase in revolutions.
__device__ __forceinline__ float sin_rev(float x) { return __builtin_amdgcn_sinf(x); }

__device__ __forceinline__ float fast_tanh(float x) {
  float xc = fminf(fmaxf(x, -10.0f), 10.0f);
  // tanh(x) = (e^{2x}-1)/(e^{2x}+1), e^{2x} = exp2(2x*log2e)
  float e = __expf(2.0f * xc);                       // -> v_exp_f32 path
  return (e - 1.0f) * __builtin_amdgcn_rcpf(e + 1.0f);
}

// ---------------------------------------------------------------------------
// Kernel 1: per-batch exclusive prefix of (300 * f0 / 24000), kept mod 1.
// The scan operator op(a,b)=frac(a+b) is associative mod 1 and keeps every
// intermediate < 1 -> full f32 precision instead of accumulating to ~3000.
// One 1024-thread block (32 wave32s) per batch.
// ---------------------------------------------------------------------------
__global__ void phase_scan_kernel(const float* __restrict__ f0,
                                  float* __restrict__ wsP,   // [BATCH][FSTRIDE]
                                  float* __restrict__ wsF) { // [BATCH][FSTRIDE]
  const int b = blockIdx.x;
  const int t = threadIdx.x;
  float f0v = (t < FRAMES) ? f0[b * FRAMES + t] : 0.0f;
  float f0n = f0v * (1.0f / 24000.0f);                 // cycles per sample
  float v   = frac1(f0v * (300.0f / 24000.0f));        // cycles per frame, mod 1

  // wave32 inclusive Hillis-Steele scan with op = frac(a+b)
  float x = v;
  const int lid = t & 31;
  #pragma unroll
  for (int d = 1; d < 32; d <<= 1) {
    float y = __shfl_up(x, d, 32);
    if (lid >= d) x = frac1(x + y);
  }

  __shared__ float wsum[32];
  const int wid = t >> 5;
  if (lid == 31) wsum[wid] = x;
  __syncthreads();
  if (wid == 0) {
    float s = wsum[lid];
    #pragma unroll
    for (int d = 1; d < 32; d <<= 1) {
      float y = __shfl_up(s, d, 32);
      if (lid >= d) s = frac1(s + y);
    }
    wsum[lid] = s;
  }
  __syncthreads();

  float off  = (wid > 0) ? wsum[wid - 1] : 0.0f;
  float incl = frac1(x + off);
  float excl = frac1(incl - v);          // exclusive prefix, mod 1

  wsP[b * FSTRIDE + t] = excl;
  wsF[b * FSTRIDE + t] = f0n;
}

// ---------------------------------------------------------------------------
// Kernel 2: harmonic source. Each wave processes groups of 16 samples.
// 9-tap dot(W, sines) done as 3 chained V_WMMA_F32_16X16X4_F32:
//   B (4x16, K x N): N = the 16 samples; lanes 0-15 hold K=0,1 in 2 VGPRs,
//                    lanes 16-31 hold K=2,3.
//   A (16x4): every row = W[4w .. 4w+3]  ->  every row of D is the dot,
//             so D[0] in each lane = result for sample (lane & 15).
// ---------------------------------------------------------------------------
__global__ void har_kernel(const float* __restrict__ wsP,
                           const float* __restrict__ wsF,
                           const float* __restrict__ W,
                           const float* __restrict__ bptr,
                           float* __restrict__ out_har,
                           float* __restrict__ out_uv) {
  // W is small + uniform -> scalar loads
  float Wk[12];
  #pragma unroll
  for (int k = 0; k < 12; ++k) Wk[k] = (k < NHARM) ? W[k] : 0.0f;
  const float bias = bptr[0];

  const int lane = threadIdx.x & 31;
  const int m    = lane & 15;                 // sample slot within group
  const bool hi  = lane >= 16;                // K=2,3 half of the wave
  const float hb = hi ? 2.0f : 0.0f;          // harmonic base offset per half

  // A matrices (row-broadcast weights), per WMMA K-block
  v2f A0, A1, A2;
  A0.x = hi ? Wk[2] : Wk[0];  A0.y = hi ? Wk[3]  : Wk[1];
  A1.x = hi ? Wk[6] : Wk[4];  A1.y = hi ? Wk[7]  : Wk[5];
  A2.x = hi ? Wk[10] : Wk[8]; A2.y = hi ? Wk[11] : Wk[9];   // Wk[9..11]=0

  const int waves_per_blk = blockDim.x >> 5;
  const int gwave  = blockIdx.x * waves_per_blk + (threadIdx.x >> 5);
  const int nwaves = gridDim.x * waves_per_blk;
  const int ngroups = TOTAL / 16;             // 300000 (never crosses a batch)

  for (int grp = gwave; grp < ngroups; grp += nwaves) {
    const int s = grp * 16 + m;               // flat sample index
    const int b = s / SAMPLES;
    const int t = s - b * SAMPLES;
    const int i = t / UPP;                    // frame
    const int j = t - i * UPP;                // offset in frame

    const float Pb  = wsP[b * FSTRIDE + i];
    const float f0n = wsF[b * FSTRIDE + i];
    const float ph  = frac1(Pb + (float)(j + 1) * f0n);   // revolutions, [0,1)
    const float uv  = (f0n > 0.0f) ? 1.0f : 0.0f;
    const float amp = SINE_AMP * uv;

    // B matrices: sines for this lane's K pair; harmonics h = 4w + hb + {1,2}
    v2f B0, B1, B2;
    B0.x = amp * sin_rev(ph * (hb + 1.0f));   // h = 1|3
    B0.y = amp * sin_rev(ph * (hb + 2.0f));   // h = 2|4
    B1.x = amp * sin_rev(ph * (hb + 5.0f));   // h = 5|7
    B1.y = amp * sin_rev(ph * (hb + 6.0f));   // h = 6|8
    B2.x = hi ? 0.0f : amp * sin_rev(ph * 9.0f); // h = 9 (10..12 zero)
    B2.y = 0.0f;

    v8f c = {};
    c = __builtin_amdgcn_wmma_f32_16x16x4_f32(false, A0, false, B0,
                                              (short)0, c, false, false);
    c = __builtin_amdgcn_wmma_f32_16x16x4_f32(false, A1, false, B1,
                                              (short)0, c, false, false);
    c = __builtin_amdgcn_wmma_f32_16x16x4_f32(false, A2, false, B2,
                                              (short)0, c, false, false);

    const float har = fast_tanh(c[0] + bias);

    // lanes 0-15 store har, lanes 16-31 store uv (both halves hold the same
    // per-sample values) -> two parallel 64B coalesced stores per wave.
    if (!hi) out_har[s] = har;
    else     out_uv[s]  = uv;
  }
}

// ---------------------------------------------------------------------------
// Kernel 3: pure streaming scale of the noise, b128 nontemporal path.
// ---------------------------------------------------------------------------
__global__ void noise_scale_kernel(const v4f* __restrict__ in,
                                   v4f* __restrict__ out, int n4) {
  const float k = SINE_AMP / 3.0f;
  const int stride = gridDim.x * blockDim.x;
  for (int idx = blockIdx.x * blockDim.x + threadIdx.x; idx < n4; idx += stride) {
    v4f v = __builtin_nontemporal_load(&in[idx]);
    v4f r; r.x = v.x * k; r.y = v.y * k; r.z = v.z * k; r.w = v.w * k;
    __builtin_nontemporal_store(r, &out[idx]);
  }
}

// ---------------------------------------------------------------------------
extern "C" void kernel_launch(void* const* d_in, const int* in_sizes, int n_in,
                              void* d_out, int out_size, void* d_ws, size_t ws_size,
                              hipStream_t stream) {
  const float* f0    = (const float*)d_in[0];   // [16,1000]
  const float* W     = (const float*)d_in[1];   // [1,9]
  const float* bias  = (const float*)d_in[2];   // [1]
  const float* noise = (const float*)d_in[3];   // [16,300000,1]
  (void)in_sizes; (void)n_in; (void)out_size; (void)ws_size;

  float* out = (float*)d_out;        // [har | noise_out | uv], TOTAL each
  float* wsP = (float*)d_ws;                    // [16][1024] phase prefixes
  float* wsF = wsP + BATCH * FSTRIDE;           // [16][1024] f0/24000

  phase_scan_kernel<<<BATCH, FSTRIDE, 0, stream>>>(f0, wsP, wsF);

  har_kernel<<<2048, 256, 0, stream>>>(wsP, wsF, W, bias,
                                       out,                 // har_source
                                       out + 2 * TOTAL);    // uv

  noise_scale_kernel<<<2048, 256, 0, stream>>>((const v4f*)noise,
                                               (v4f*)(out + TOTAL),
                                               TOTAL / 4);
}